// BinnedSelectKnnModule_62723702390789
// MI455X (gfx1250) — compile-verified
//
#include <hip/hip_runtime.h>

// ---------------------------------------------------------------------------
// Segmented exact kNN for MI455X (gfx1250, wave32).
//   B=8 segments x S=4096 points x D=4, K=64.
//
//  * Block (256 thr / 8 waves) async-stages its whole segment into LDS via
//    GLOBAL_LOAD_ASYNC_TO_LDS_B128 (+ s_wait_asynccnt), then computes norms.
//  * Gram tiles via V_WMMA_F32_16X16X4_F32: D=4 == WMMA K=4, one instruction
//    per 16x16 tile of (-2 * X * X^T), fp32 exact.
//  * WMMA C-layout -> per-row lane layout with a single xor-16 DS_SWIZZLE.
//  * Selection carries packed u64 keys (dist_bits<<32 | idx): tie-free,
//    matches top_k ordering; each bitonic stage = 2x ds_swizzle +
//    v_cmp_lt_u64 + s_xor(dir mask) + 2x v_cndmask.
//  * Scalar-branch threshold early-out: a 32-candidate batch is merged only
//    if ballot(candKey < 64th-best key) != 0 (exact; skips ~90% of merges:
//    E[insertions] = 64*ln(S/64) ~ 266 of 4096).
// ---------------------------------------------------------------------------

typedef __attribute__((ext_vector_type(2))) float v2f;
typedef __attribute__((ext_vector_type(8))) float v8f;
typedef unsigned long long u64;
typedef unsigned int u32;

#define KNN_K 64
#define ROWS_PER_BLOCK 128
#define NWAVES 8

// ----- immediate xor-swizzle across wave32 ---------------------------------

template <int XORMASK>
__device__ __forceinline__ float swz_f(float v) {
  constexpr int ctl = (XORMASK << 5) | 0x1f;  // and=0x1f, or=0, xor=XORMASK
  return __int_as_float(__builtin_amdgcn_ds_swizzle(__float_as_int(v), ctl));
}
template <int XORMASK>
__device__ __forceinline__ u64 swz_k(u64 k) {
  constexpr int ctl = (XORMASK << 5) | 0x1f;
  u32 lo = (u32)__builtin_amdgcn_ds_swizzle((int)(u32)k, ctl);
  u32 hi = (u32)__builtin_amdgcn_ds_swizzle((int)(u32)(k >> 32), ctl);
  return ((u64)hi << 32) | lo;
}

// Bitonic compare-exchange stages on unique u64 keys.
// take = (other < mine) XOR flip  -> v_cmp_lt_u64 + s_xor + cndmask pair.
template <int KBLK, int J>
__device__ __forceinline__ void sortStage(u64& k, int lane) {
  u64 ok = swz_k<J>(k);
  bool flip = ((lane & J) != 0) != ((lane & KBLK) != 0);  // keep-max lanes
  if ((ok < k) != flip) k = ok;
}

template <int J>
__device__ __forceinline__ void mergeStage(u64& k, int lane) {
  u64 ok = swz_k<J>(k);
  bool flip = (lane & J) != 0;
  if ((ok < k) != flip) k = ok;
}

// Full ascending bitonic sort of 32 keys, one per lane.
__device__ __forceinline__ void sort32(u64& k, int lane) {
  sortStage<2, 1>(k, lane);
  sortStage<4, 2>(k, lane);
  sortStage<4, 1>(k, lane);
  sortStage<8, 4>(k, lane);
  sortStage<8, 2>(k, lane);
  sortStage<8, 1>(k, lane);
  sortStage<16, 8>(k, lane);
  sortStage<16, 4>(k, lane);
  sortStage<16, 2>(k, lane);
  sortStage<16, 1>(k, lane);
  mergeStage<16>(k, lane);  // k=32 block is ascending for all lanes
  mergeStage<8>(k, lane);
  mergeStage<4>(k, lane);
  mergeStage<2>(k, lane);
  mergeStage<1>(k, lane);
}

// Merge fresh 32 keys into running ascending top-64
// (r0 = elems 0..31 by lane, r1 = elems 32..63 by lane).
__device__ __forceinline__ void mergeTop64(u64& r0, u64& r1, u64 s, int lane) {
  sort32(s, lane);
  u64 rs = swz_k<31>(s);       // lane reversal -> descending
  if (rs < r1) r1 = rs;        // keep-64-of-128
  if (r1 < r0) { u64 t = r0; r0 = r1; r1 = t; }  // bitonic stage d=32 (local)
  mergeStage<16>(r0, lane); mergeStage<16>(r1, lane);
  mergeStage<8>(r0, lane);  mergeStage<8>(r1, lane);
  mergeStage<4>(r0, lane);  mergeStage<4>(r1, lane);
  mergeStage<2>(r0, lane);  mergeStage<2>(r1, lane);
  mergeStage<1>(r0, lane);  mergeStage<1>(r1, lane);
}

// 64th-best key = element 63 = upper-half key on lane 31 (scalar result).
__device__ __forceinline__ u64 kthBest(u64 r1) {
  u32 lo = (u32)__builtin_amdgcn_readlane((int)(u32)r1, 31);
  u32 hi = (u32)__builtin_amdgcn_readlane((int)(u32)(r1 >> 32), 31);
  return ((u64)hi << 32) | lo;
}

__device__ __forceinline__ u64 packKey(float d, int idx) {
  return ((u64)(u32)__float_as_int(d) << 32) | (u32)idx;  // d >= 0
}

// ----- kernel ---------------------------------------------------------------

__global__ __launch_bounds__(32 * NWAVES)
void BinnedSelectKnn_kernel(const float* __restrict__ coords,
                            int* __restrict__ out_idx,
                            float* __restrict__ out_dist,
                            int S) {
  extern __shared__ __align__(16) char smem[];
  float4* pts = (float4*)smem;       // [S] segment coordinates
  float*  sqn = (float*)(pts + S);   // [S] squared norms

  const int tid  = threadIdx.x;
  const int lane = tid & 31;
  const int wave = tid >> 5;

  const int blocksPerSeg = S / ROWS_PER_BLOCK;
  const int seg        = blockIdx.x / blocksPerSeg;
  const int segBase    = seg * S;
  const int qBlockBase = (blockIdx.x % blocksPerSeg) * ROWS_PER_BLOCK;

  // --- Async-stage the segment into LDS (CDNA5 async global->LDS path) ---
  {
    const u64 gbase = (u64)coords + (u64)segBase * 16ull;
    for (int p = tid; p < S; p += blockDim.x) {
      unsigned ldsAddr = (unsigned)(size_t)(pts + p);  // low 32 bits = LDS addr
      unsigned gOff    = (unsigned)p * 16u;
      asm volatile("global_load_async_to_lds_b128 %0, %1, %2 offset:0"
                   :: "v"(ldsAddr), "v"(gOff), "s"(gbase) : "memory");
    }
    asm volatile("s_wait_asynccnt 0x0" ::: "memory");
  }
  __syncthreads();
  for (int p = tid; p < S; p += blockDim.x) {
    float4 v = pts[p];
    sqn[p] = v.x * v.x + v.y * v.y + v.z * v.z + v.w * v.w;
  }
  __syncthreads();

  // --- Per-wave 16-row query tile ---
  const int  qw     = qBlockBase + wave * 16;  // segment-local first row
  const bool hiHalf = lane >= 16;
  const int  l15    = lane & 15;

  // A fragment = -2 * queries. 16x4 f32 A layout (ISA 7.12.2):
  //   lanes 0-15:  M=lane,    VGPR0=K0, VGPR1=K1
  //   lanes 16-31: M=lane-16, VGPR0=K2, VGPR1=K3
  float4 qa = pts[qw + l15];
  v2f afrag;
  afrag.x = -2.0f * (hiHalf ? qa.z : qa.x);
  afrag.y = -2.0f * (hiHalf ? qa.w : qa.y);

  // C/D layout: VGPR r = row r on lanes 0-15, row r+8 on lanes 16-31.
  const int rowOff = hiHalf ? 8 : 0;
  float sqm[8];
#pragma unroll
  for (int r = 0; r < 8; ++r) sqm[r] = sqn[qw + rowOff + r];

  // Running sorted top-64 per row (2 keys per lane, 16 rows) + thresholds.
  u64 R0[16], R1[16], thr[16];
#pragma unroll
  for (int r = 0; r < 16; ++r) {
    R0[r] = ~0ull; R1[r] = ~0ull; thr[r] = ~0ull;
  }

  // --- Sweep candidate columns, 32 at a time (two WMMA N-tiles) ---
  for (int cb = 0; cb < S; cb += 32) {
    float4 p0 = pts[cb + l15];
    float4 p1 = pts[cb + 16 + l15];
    v2f b0, b1;
    b0.x = hiHalf ? p0.z : p0.x;  b0.y = hiHalf ? p0.w : p0.y;
    b1.x = hiHalf ? p1.z : p1.x;  b1.y = hiHalf ? p1.w : p1.y;

    v8f c0 = {}, c1 = {};
    c0 = __builtin_amdgcn_wmma_f32_16x16x4_f32(false, afrag, false, b0,
                                               (short)0, c0, false, false);
    c1 = __builtin_amdgcn_wmma_f32_16x16x4_f32(false, afrag, false, b1,
                                               (short)0, c1, false, false);

    const float sq0 = sqn[cb + l15];        // same col for both lane halves
    const float sq1 = sqn[cb + 16 + l15];
    const int   si  = segBase + cb + lane;  // candidate index for both rows

#pragma unroll
    for (int r = 0; r < 8; ++r) {
      float d0 = fmaxf(c0[r] + sqm[r] + sq0, 0.0f);  // rows r / r+8, cols 0-15
      float d1 = fmaxf(c1[r] + sqm[r] + sq1, 0.0f);  // rows r / r+8, cols 16-31
      // Redistribute so lane L holds column cb+L of one row:
      float d0x = swz_f<16>(d0);
      float d1x = swz_f<16>(d1);
      u64 candLo = packKey(hiHalf ? d1x : d0, si);  // row qw + r
      u64 candHi = packKey(hiHalf ? d1 : d0x, si);  // row qw + r + 8

      // Exact early-out: only merge a batch that can change the top-64.
      if (__builtin_amdgcn_ballot_w32(candLo < thr[r])) {
        mergeTop64(R0[r], R1[r], candLo, lane);
        thr[r] = kthBest(R1[r]);
      }
      if (__builtin_amdgcn_ballot_w32(candHi < thr[r + 8])) {
        mergeTop64(R0[r + 8], R1[r + 8], candHi, lane);
        thr[r + 8] = kthBest(R1[r + 8]);
      }
    }
  }

  // --- Emit sorted top-64 (idx block first, then dist block) ---
#pragma unroll
  for (int r = 0; r < 16; ++r) {
    const int q = segBase + qw + r;  // global query row
    out_idx [q * KNN_K + lane]      = (int)(u32)R0[r];
    out_idx [q * KNN_K + 32 + lane] = (int)(u32)R1[r];
    out_dist[q * KNN_K + lane]      = __int_as_float((int)(u32)(R0[r] >> 32));
    out_dist[q * KNN_K + 32 + lane] = __int_as_float((int)(u32)(R1[r] >> 32));
  }
}

// ----- host launch ----------------------------------------------------------

extern "C" void kernel_launch(void* const* d_in, const int* in_sizes, int n_in,
                              void* d_out, int out_size, void* d_ws, size_t ws_size,
                              hipStream_t stream) {
  // inputs: [0]=K (scalar, assumed 64), [1]=coordinates f32 [N*4],
  //         [2]=row_splits i32 [B+1] (equal segments)
  const float* coords = (const float*)d_in[1];
  const int B = in_sizes[2] - 1;
  const int N = in_sizes[1] / 4;
  const int S = N / B;

  int*   out_idx  = (int*)d_out;
  float* out_dist = (float*)d_out + (size_t)N * KNN_K;

  dim3 grid(N / ROWS_PER_BLOCK);
  dim3 block(32 * NWAVES);
  size_t shmem = (size_t)S * sizeof(float4)   // coords
               + (size_t)S * sizeof(float);   // norms

  BinnedSelectKnn_kernel<<<grid, block, shmem, stream>>>(coords, out_idx,
                                                         out_dist, S);
}